// VimBlock_5059471475435
// MI455X (gfx1250) — compile-verified
//
#include <hip/hip_runtime.h>
#include <hip/hip_bf16.h>
#include <math.h>

// ---------------------------------------------------------------------------
// Vim block (bidirectional Mamba) for MI455X / gfx1250, fp32 throughout.
// GEMMs: V_WMMA_F32_16X16X4_F32, 16x64 strip per wave (4 accumulators).
// Scan: 3-pass chunked parallel scan (32 chunks x 128) to cut the serial
// critical path 4096 -> 288 dependent steps with 32x more wave parallelism.
// ---------------------------------------------------------------------------

#define BATCH   2
#define DIMC    192          // model dim
#define EINNER  384          // D_INNER
#define NSTATE  16
#define RDT     12           // DT_RANK
#define XDBL    44           // R + 2*N
#define SEQL    4096         // H*W
#define DCONV   4
#define EPSLN   1e-5f
#define NCHUNK  32
#define CLEN    (SEQL / NCHUNK)   // 128

typedef __attribute__((ext_vector_type(2))) float v2f;
typedef __attribute__((ext_vector_type(8))) float v8f;

__device__ __forceinline__ float silu_f(float x) { return x / (1.0f + __expf(-x)); }
__device__ __forceinline__ float softplus_f(float x) {
  return (x > 20.0f) ? x : log1pf(__expf(x));
}

// ---------------------------------------------------------------------------
// Workspace layout (floats). Total ~= 32.8M floats ~= 131 MiB.
// ---------------------------------------------------------------------------
#define OFF_XSEQ   0u                               // [B][L][DIMC]
#define OFF_HN     (OFF_XSEQ + BATCH*SEQL*DIMC)     // [B][L][DIMC]
#define OFF_XZ     (OFF_HN   + BATCH*SEQL*DIMC)     // [B][L][2*EINNER]
#define OFF_XC     (OFF_XZ   + BATCH*SEQL*2*EINNER) // [dir][B][L][EINNER]
#define OFF_DELTA  (OFF_XC   + 2*BATCH*SEQL*EINNER) // [dir][B][L][EINNER]
#define OFF_BS     (OFF_DELTA+ 2*BATCH*SEQL*EINNER) // [dir][B][L][NSTATE]
#define OFF_CS     (OFF_BS   + 2*BATCH*SEQL*NSTATE) // [dir][B][L][NSTATE]
#define OFF_YDIR   (OFF_CS   + 2*BATCH*SEQL*NSTATE) // [dir][B][L][EINNER]
#define OFF_YCOMB  (OFF_YDIR + 2*BATCH*SEQL*EINNER) // [B][L][EINNER]
#define NCHAN      (2*BATCH*EINNER*NSTATE)          // 24576 channel-states
#define OFF_HLOC   (OFF_YCOMB+ BATCH*SEQL*EINNER)   // [chan][NCHUNK]
#define OFF_PRODA  (OFF_HLOC + NCHAN*NCHUNK)        // [chan][NCHUNK]
#define OFF_H0     (OFF_PRODA+ NCHAN*NCHUNK)        // [chan][NCHUNK]

// ---------------------------------------------------------------------------
// K1: x_seq = reshape(x) + pos_emb;  h = LayerNorm(x_seq)*g + b
// ---------------------------------------------------------------------------
__global__ __launch_bounds__(DIMC) void k_norm(
    const float* __restrict__ x,        // (B, DIMC, L)
    const float* __restrict__ pos_emb,  // (1, L, DIMC)
    const float* __restrict__ g,
    const float* __restrict__ bb,
    float* __restrict__ ws) {
  __shared__ float2 red[256];
  const int tok = blockIdx.x;               // b*L + l
  const int b = tok >> 12, l = tok & (SEQL - 1);
  const int c = threadIdx.x;

  float v = x[(size_t)b * DIMC * SEQL + (size_t)c * SEQL + l] +
            pos_emb[(size_t)l * DIMC + c];

  red[c] = make_float2(v, v * v);
  if (c < 256 - DIMC) red[DIMC + c] = make_float2(0.f, 0.f);
  __syncthreads();
  for (int s = 128; s >= 1; s >>= 1) {
    if (c < s) {
      float2 a = red[c], o = red[c + s];
      red[c] = make_float2(a.x + o.x, a.y + o.y);
    }
    __syncthreads();
  }
  const float mu  = red[0].x * (1.0f / DIMC);
  const float var = red[0].y * (1.0f / DIMC) - mu * mu;
  const float rs  = rsqrtf(var + EPSLN);

  float* xseq = ws + OFF_XSEQ;
  float* hn   = ws + OFF_HN;
  xseq[(size_t)tok * DIMC + c] = v;
  hn  [(size_t)tok * DIMC + c] = (v - mu) * rs * g[c] + bb[c];
}

// ---------------------------------------------------------------------------
// fp32 WMMA GEMM: one wave computes a 16x64 strip (4 accumulators), so each
// A fragment feeds 4 WMMAs. A: (M x K) row-major, W: (K x N) row-major.
// A 16x4 frag: lanes 0..15 -> K=k0,k0+1 ; lanes 16..31 -> K=k0+2,k0+3
// B 4x16 frag: vgpr0 = rows {k0, k0+2}, vgpr1 = rows {k0+1, k0+3}
// ---------------------------------------------------------------------------
template <int K, int N>
__device__ __forceinline__ void gemm_strip_16x64(
    const float* __restrict__ A, const float* __restrict__ W,
    int mt, int nt, int lane, v8f acc[4]) {
  const int half = lane >> 4;
  const int l15  = lane & 15;
  const float* arow = A + (size_t)(mt + l15) * K;

  for (int k0 = 0; k0 < K; k0 += 4) {
    const int ka = k0 + half * 2;
    v2f af;
    af.x = arow[ka];
    af.y = arow[ka + 1];
    if (k0 + 8 < K) {
      // locality 3 -> WGP-scope global_prefetch_b8 (pulls into near caches);
      // operands are L2-resident so this warms the WGP cache for the stream.
      __builtin_prefetch(&arow[k0 + 8], 0, 3);
      __builtin_prefetch(&W[(size_t)(k0 + 8) * N + nt + l15], 0, 3);
    }
#pragma unroll
    for (int j = 0; j < 4; ++j) {
      v2f bf;
      bf.x = W[(size_t)ka * N + nt + j * 16 + l15];
      bf.y = W[(size_t)(ka + 1) * N + nt + j * 16 + l15];
      acc[j] = __builtin_amdgcn_wmma_f32_16x16x4_f32(false, af, false, bf,
                                                     (short)0, acc[j],
                                                     false, false);
    }
  }
}

// K2: xz = h @ in_proj_w    (M=8192, K=192, N=768)
__global__ __launch_bounds__(32) void k_gemm_inproj(
    const float* __restrict__ A,   // h_norm  (M x 192)
    const float* __restrict__ W,   // in_proj (192 x 768)
    float* __restrict__ C) {       // xz      (M x 768)
  const int mt = blockIdx.x * 16;
  const int nt = blockIdx.y * 64;
  const int lane = threadIdx.x;
  const int half = lane >> 4;
  const int l15  = lane & 15;

  v8f acc[4] = {};
  gemm_strip_16x64<DIMC, 2 * EINNER>(A, W, mt, nt, lane, acc);

#pragma unroll
  for (int j = 0; j < 4; ++j)
#pragma unroll
    for (int i = 0; i < 8; ++i)
      C[(size_t)(mt + half * 8 + i) * (2 * EINNER) + nt + j * 16 + l15] =
          acc[j][i];
}

// ---------------------------------------------------------------------------
// K3: causal depthwise conv (width 4) + SiLU, per direction.
// ---------------------------------------------------------------------------
__global__ __launch_bounds__(EINNER) void k_conv(
    const float* __restrict__ ws_xz,
    const float* __restrict__ fwd_w, const float* __restrict__ fwd_b,
    const float* __restrict__ bwd_w, const float* __restrict__ bwd_b,
    float* __restrict__ ws_xc) {
  const int tok = blockIdx.x;
  const int dir = blockIdx.y;
  const int b = tok >> 12, l = tok & (SEQL - 1);
  const int e = threadIdx.x;

  const float* w  = (dir == 0) ? fwd_w : bwd_w;
  const float* bi = (dir == 0) ? fwd_b : bwd_b;

  float acc = bi[e];
#pragma unroll
  for (int j = 0; j < DCONV; ++j) {
    const int t = l - (DCONV - 1) + j;
    if (t >= 0) {
      const int ts = (dir == 0) ? t : (SEQL - 1 - t);   // flipped read for bwd
      acc += w[e * DCONV + j] *
             ws_xz[((size_t)b * SEQL + ts) * (2 * EINNER) + e];
    }
  }
  ws_xc[(((size_t)dir * BATCH + b) * SEQL + l) * EINNER + e] = silu_f(acc);
}

// ---------------------------------------------------------------------------
// K4: x_dbl = xc @ xproj_w (384->44); delta = softplus(x_dbl[:R]@dt_w + dt_b);
//     B_ssm, C_ssm extracted.
// ---------------------------------------------------------------------------
__global__ __launch_bounds__(128) void k_xproj(
    const float* __restrict__ ws_xc,
    const float* __restrict__ fw_xp, const float* __restrict__ fw_dtw,
    const float* __restrict__ fw_dtb,
    const float* __restrict__ bw_xp, const float* __restrict__ bw_dtw,
    const float* __restrict__ bw_dtb,
    float* __restrict__ ws_delta, float* __restrict__ ws_B,
    float* __restrict__ ws_C) {
  __shared__ float sx[EINNER];
  __shared__ float sdbl[XDBL];
  const int tok = blockIdx.x;
  const int dir = blockIdx.y;
  const int tid = threadIdx.x;
  const size_t base = ((size_t)dir * BATCH * SEQL + tok);

  const float* xp  = (dir == 0) ? fw_xp  : bw_xp;
  const float* dtw = (dir == 0) ? fw_dtw : bw_dtw;
  const float* dtb = (dir == 0) ? fw_dtb : bw_dtb;

  for (int e = tid; e < EINNER; e += 128)
    sx[e] = ws_xc[base * EINNER + e];
  __syncthreads();

  if (tid < XDBL) {
    float acc = 0.f;
    for (int e = 0; e < EINNER; ++e) acc += sx[e] * xp[e * XDBL + tid];
    sdbl[tid] = acc;
  }
  __syncthreads();

  for (int e = tid; e < EINNER; e += 128) {
    float d = dtb[e];
#pragma unroll
    for (int r = 0; r < RDT; ++r) d += sdbl[r] * dtw[r * EINNER + e];
    ws_delta[base * EINNER + e] = softplus_f(d);
  }
  if (tid < NSTATE) {
    ws_B[base * NSTATE + tid] = sdbl[RDT + tid];
    ws_C[base * NSTATE + tid] = sdbl[RDT + NSTATE + tid];
  }
}

// ---------------------------------------------------------------------------
// Chunked parallel scan. Channel-state (dir,b,e,n) lives on one lane:
// lane = ch_half*16 + n, wave handles 2 channels.
// grid (EINNER/16, NCHUNK, BATCH*2), block 256 (8 waves = 16 channels).
// ---------------------------------------------------------------------------
// Pass 1: local scan per chunk with h0 = 0; record (h_local, prod_a).
__global__ __launch_bounds__(256) void k_scan_pass1(
    const float* __restrict__ ws_xc, const float* __restrict__ ws_delta,
    const float* __restrict__ ws_B,
    const float* __restrict__ fw_Alog, const float* __restrict__ bw_Alog,
    float* __restrict__ ws_hloc, float* __restrict__ ws_proda) {
  const int dir   = blockIdx.z >> 1;
  const int b     = blockIdx.z & 1;
  const int chunk = blockIdx.y;
  const int wave  = threadIdx.x >> 5;
  const int lane  = threadIdx.x & 31;
  const int n     = lane & 15;
  const int e     = blockIdx.x * 16 + wave * 2 + (lane >> 4);

  const float* Alog = (dir == 0) ? fw_Alog : bw_Alog;
  const float Aen = -__expf(Alog[e * NSTATE + n]);

  const size_t db = ((size_t)dir * BATCH + b) * SEQL;
  float h = 0.0f, prod = 1.0f;

  const int t0 = chunk * CLEN;
  for (int t = t0; t < t0 + CLEN; ++t) {
    const size_t te = (db + t) * EINNER + e;
    const size_t tn = (db + t) * NSTATE + n;
    const float dlt = ws_delta[te];
    const float xv  = ws_xc[te];
    const float Bt  = ws_B[tn];
    const float dA  = __expf(dlt * Aen);
    h = dA * h + (dlt * xv) * Bt;
    prod *= dA;
  }
  const size_t ci = (((size_t)dir * BATCH + b) * EINNER + e) * NSTATE + n;
  ws_hloc [ci * NCHUNK + chunk] = h;
  ws_proda[ci * NCHUNK + chunk] = prod;
}

// Pass 2: carry propagation across the 32 chunks of each channel-state.
__global__ __launch_bounds__(256) void k_scan_fix(
    const float* __restrict__ ws_hloc, const float* __restrict__ ws_proda,
    float* __restrict__ ws_h0) {
  const size_t ci = (size_t)blockIdx.x * 256 + threadIdx.x;  // < NCHAN
  float carry = 0.0f;
#pragma unroll
  for (int c = 0; c < NCHUNK; ++c) {
    ws_h0[ci * NCHUNK + c] = carry;
    carry = ws_proda[ci * NCHUNK + c] * carry + ws_hloc[ci * NCHUNK + c];
  }
}

// Pass 3: replay chunk seeded with true incoming state; emit y.
__global__ __launch_bounds__(256) void k_scan_pass3(
    const float* __restrict__ ws_xc, const float* __restrict__ ws_delta,
    const float* __restrict__ ws_B,  const float* __restrict__ ws_C,
    const float* __restrict__ fw_Alog, const float* __restrict__ fw_D,
    const float* __restrict__ bw_Alog, const float* __restrict__ bw_D,
    const float* __restrict__ ws_h0, float* __restrict__ ws_y) {
  const int dir   = blockIdx.z >> 1;
  const int b     = blockIdx.z & 1;
  const int chunk = blockIdx.y;
  const int wave  = threadIdx.x >> 5;
  const int lane  = threadIdx.x & 31;
  const int n     = lane & 15;
  const int e     = blockIdx.x * 16 + wave * 2 + (lane >> 4);

  const float* Alog = (dir == 0) ? fw_Alog : bw_Alog;
  const float* Dp   = (dir == 0) ? fw_D    : bw_D;
  const float Aen = -__expf(Alog[e * NSTATE + n]);
  const float De  = Dp[e];

  const size_t db = ((size_t)dir * BATCH + b) * SEQL;
  const size_t ci = (((size_t)dir * BATCH + b) * EINNER + e) * NSTATE + n;
  float h = ws_h0[ci * NCHUNK + chunk];

  const int t0 = chunk * CLEN;
  for (int t = t0; t < t0 + CLEN; ++t) {
    const size_t te = (db + t) * EINNER + e;
    const size_t tn = (db + t) * NSTATE + n;
    const float dlt = ws_delta[te];
    const float xv  = ws_xc[te];
    const float Bt  = ws_B[tn];
    const float Ct  = ws_C[tn];

    const float dA = __expf(dlt * Aen);
    h = dA * h + (dlt * xv) * Bt;

    float p = h * Ct;
    // reduce over n (stays within each 16-lane half)
    p += __shfl_xor(p, 8, 32);
    p += __shfl_xor(p, 4, 32);
    p += __shfl_xor(p, 2, 32);
    p += __shfl_xor(p, 1, 32);
    if (n == 0) ws_y[te] = p + De * xv;
  }
}

// ---------------------------------------------------------------------------
// K6: y = (y_fwd[l] + y_bwd[L-1-l]) * silu(z)
// ---------------------------------------------------------------------------
__global__ __launch_bounds__(EINNER) void k_combine(
    const float* __restrict__ ws_y, const float* __restrict__ ws_xz,
    float* __restrict__ ws_yc) {
  const int tok = blockIdx.x;
  const int b = tok >> 12, l = tok & (SEQL - 1);
  const int e = threadIdx.x;

  const float yf = ws_y[(((size_t)0 * BATCH + b) * SEQL + l) * EINNER + e];
  const float yb = ws_y[(((size_t)1 * BATCH + b) * SEQL + (SEQL - 1 - l)) * EINNER + e];
  const float z  = ws_xz[(size_t)tok * (2 * EINNER) + EINNER + e];
  ws_yc[(size_t)tok * EINNER + e] = (yf + yb) * silu_f(z);
}

// ---------------------------------------------------------------------------
// K7: out = x_seq + y @ out_proj_w  (M=8192, K=384, N=192), fp32 WMMA,
// stored transposed into (B, DIMC, L) output layout.
// ---------------------------------------------------------------------------
__global__ __launch_bounds__(32) void k_gemm_outproj(
    const float* __restrict__ A,     // y_comb (M x 384)
    const float* __restrict__ W,     // out_proj (384 x 192)
    const float* __restrict__ xseq,  // residual (M x 192)
    float* __restrict__ out) {       // (B, DIMC, L)
  const int mt = blockIdx.x * 16;
  const int nt = blockIdx.y * 64;
  const int lane = threadIdx.x;
  const int half = lane >> 4;
  const int l15  = lane & 15;

  v8f acc[4] = {};
  gemm_strip_16x64<EINNER, DIMC>(A, W, mt, nt, lane, acc);

#pragma unroll
  for (int j = 0; j < 4; ++j) {
    const int c = nt + j * 16 + l15;
#pragma unroll
    for (int i = 0; i < 8; ++i) {
      const int m = mt + half * 8 + i;        // token index (tiles never cross b)
      const int b = m >> 12, l = m & (SEQL - 1);
      out[(size_t)b * DIMC * SEQL + (size_t)c * SEQL + l] =
          acc[j][i] + xseq[(size_t)m * DIMC + c];
    }
  }
}

// ---------------------------------------------------------------------------
extern "C" void kernel_launch(void* const* d_in, const int* in_sizes, int n_in,
                              void* d_out, int out_size, void* d_ws, size_t ws_size,
                              hipStream_t stream) {
  const float* x        = (const float*)d_in[0];
  const float* pos_emb  = (const float*)d_in[1];
  const float* norm_g   = (const float*)d_in[2];
  const float* norm_b   = (const float*)d_in[3];
  const float* in_proj  = (const float*)d_in[4];
  const float* fw_cw    = (const float*)d_in[5];
  const float* fw_cb    = (const float*)d_in[6];
  const float* fw_xp    = (const float*)d_in[7];
  const float* fw_dtw   = (const float*)d_in[8];
  const float* fw_dtb   = (const float*)d_in[9];
  const float* fw_Alog  = (const float*)d_in[10];
  const float* fw_D     = (const float*)d_in[11];
  const float* bw_cw    = (const float*)d_in[12];
  const float* bw_cb    = (const float*)d_in[13];
  const float* bw_xp    = (const float*)d_in[14];
  const float* bw_dtw   = (const float*)d_in[15];
  const float* bw_dtb   = (const float*)d_in[16];
  const float* bw_Alog  = (const float*)d_in[17];
  const float* bw_D     = (const float*)d_in[18];
  const float* out_proj = (const float*)d_in[19];

  float* ws  = (float*)d_ws;
  float* out = (float*)d_out;

  const int NTOK = BATCH * SEQL;  // 8192

  // 1) pos-emb + layernorm
  k_norm<<<NTOK, DIMC, 0, stream>>>(x, pos_emb, norm_g, norm_b, ws);

  // 2) xz = h @ in_proj_w   (16x64 strips: grid 512 x 12)
  k_gemm_inproj<<<dim3(NTOK / 16, (2 * EINNER) / 64), 32, 0, stream>>>(
      ws + OFF_HN, in_proj, ws + OFF_XZ);

  // 3) depthwise causal conv + SiLU (both directions)
  k_conv<<<dim3(NTOK, 2), EINNER, 0, stream>>>(
      ws + OFF_XZ, fw_cw, fw_cb, bw_cw, bw_cb, ws + OFF_XC);

  // 4) x-proj, delta, B, C
  k_xproj<<<dim3(NTOK, 2), 128, 0, stream>>>(
      ws + OFF_XC, fw_xp, fw_dtw, fw_dtb, bw_xp, bw_dtw, bw_dtb,
      ws + OFF_DELTA, ws + OFF_BS, ws + OFF_CS);

  // 5) chunked parallel scan (3 passes)
  dim3 sg(EINNER / 16, NCHUNK, BATCH * 2);
  k_scan_pass1<<<sg, 256, 0, stream>>>(
      ws + OFF_XC, ws + OFF_DELTA, ws + OFF_BS,
      fw_Alog, bw_Alog, ws + OFF_HLOC, ws + OFF_PRODA);
  k_scan_fix<<<NCHAN / 256, 256, 0, stream>>>(
      ws + OFF_HLOC, ws + OFF_PRODA, ws + OFF_H0);
  k_scan_pass3<<<sg, 256, 0, stream>>>(
      ws + OFF_XC, ws + OFF_DELTA, ws + OFF_BS, ws + OFF_CS,
      fw_Alog, fw_D, bw_Alog, bw_D, ws + OFF_H0, ws + OFF_YDIR);

  // 6) combine directions + gate
  k_combine<<<NTOK, EINNER, 0, stream>>>(ws + OFF_YDIR, ws + OFF_XZ,
                                         ws + OFF_YCOMB);

  // 7) out-proj + residual, transposed store (grid 512 x 3)
  k_gemm_outproj<<<dim3(NTOK / 16, DIMC / 64), 32, 0, stream>>>(
      ws + OFF_YCOMB, out_proj, ws + OFF_XSEQ, out);
}